// BoxDetectionLoss_34411277975623
// MI455X (gfx1250) — compile-verified
//
#include <hip/hip_runtime.h>
#include <hip/hip_bf16.h>

typedef __attribute__((ext_vector_type(2))) float v2f;
typedef __attribute__((ext_vector_type(8))) float v8f;

#define BB 8
#define AA 3
#define HH 512
#define WW 512
#define TT 16
static constexpr int PLANE   = HH * WW;       // 262144 floats per channel plane
static constexpr int CSTRIDE = 9 * PLANE;     // per-batch stride (9 channels)

__device__ __forceinline__ float sigmoidf_fast(float x) {
    return 1.0f / (1.0f + __expf(-x));
}

// Lane-XOR add via ds_swizzle_b32 (group-of-32 mode, pattern in immediate:
// xor_mask[14:10] | or_mask[9:5] | and_mask[4:0]). No address setup needed,
// unlike ds_bpermute-based __shfl_xor.
#define XOR_ADD(v, imm)                                                        \
    (v) += __int_as_float(__builtin_amdgcn_ds_swizzle(__float_as_int(v), imm))

// Exact f32 wave-wide sum using V_WMMA_F32_16X16X4_F32 as a 32->16 reduction,
// finished by 4 xor-swizzles. A = all-ones 16x4, B = {s, 0}: every column j of
// D accumulates s_j + s_{j+16} (all-ones A makes the exact K-row layout of the
// B VGPRs irrelevant). Requires EXEC == all ones (call from convergent code).
__device__ __forceinline__ float wave_sum32(float s) {
    v2f a; a.x = 1.0f; a.y = 1.0f;
    v2f b; b.x = s;    b.y = 0.0f;
    v8f c = {};
    c = __builtin_amdgcn_wmma_f32_16x16x4_f32(false, a, false, b,
                                              (short)0, c, false, false);
    float v = c[0];            // colsum[lane % 16], replicated across rows
    XOR_ADD(v, 0x041F);        // SWAPX1
    XOR_ADD(v, 0x081F);        // SWAPX2
    XOR_ADD(v, 0x101F);        // SWAPX4
    XOR_ADD(v, 0x201F);        // SWAPX8
    return v;                  // total of all 32 lanes, in every lane
}

// Stage 1: each block handles (b, a, 2 rows). 256 threads, thread -> float4 of
// 4 consecutive columns in one row. Block-uniform fast path skips the delta
// planes entirely when no target box lands on this row pair.
__global__ __launch_bounds__(256) void box_loss_stage1(
        const float* __restrict__ pol,
        const int*   __restrict__ boxes,
        const float* __restrict__ probs,
        float*       __restrict__ partial) {
    const int blk    = blockIdx.x;
    const int rowblk = blk & 255;           // 256 row-pairs
    const int a      = (blk >> 8) % 3;
    const int b      = blk / 768;
    const int tid    = threadIdx.x;
    const int r      = (rowblk << 1) | (tid >> 7);
    const int c0     = (tid & 127) << 2;

    const float* base = pol + (size_t)b * CSTRIDE + (size_t)(a * 3) * PLANE
                            + (size_t)r * WW + c0;

    // Block-uniform target data (b is uniform) -> scalar loads / SGPRs.
    const int*   tb  = boxes + b * TT * 4;
    const float* tpb = probs + b * TT;
    unsigned long long bkey[TT];
    float tprob[TT];
    bool rowHit = false;
#pragma unroll
    for (int t = 0; t < TT; ++t) {
        const int br  = tb[4 * t + 0];
        const int bc  = tb[4 * t + 1];
        const int br2 = tb[4 * t + 2];
        const int bc2 = tb[4 * t + 3];
        bkey[t] = ((unsigned long long)(unsigned)(br * WW + bc) << 32)
                | (unsigned)(br2 * WW + bc2);
        tprob[t] = tpb[t];
        rowHit |= (((unsigned)br >> 1) == (unsigned)rowblk);
    }

    const float4 vconf = *(const float4*)(base + 2 * (size_t)PLANE);
    float sum = 0.0f;

    if (!rowHit) {
        // Fast path (~94% of blocks): per_anchor = conf^2 only.
#pragma unroll
        for (int k = 0; k < 4; ++k) {
            const float x = (&vconf.x)[k];
            const float conf = sigmoidf_fast(x);
            sum += conf * conf;
        }
    } else {
        const float4 vr = *(const float4*)(base);
        const float4 vc = *(const float4*)(base + (size_t)PLANE);
#pragma unroll
        for (int k = 0; k < 4; ++k) {
            const float dr   = sigmoidf_fast((&vr.x)[k]) * 9.0f;
            const float dc   = sigmoidf_fast((&vc.x)[k]) * 16.0f;
            const float conf = sigmoidf_fast((&vconf.x)[k]);
            const float pr2  = fminf((float)r + dr, 511.0f);
            const float pc2  = fminf((float)(c0 + k) + dc, 511.0f);
            const float prr  = rintf(pr2);   // round-half-even == jnp.round
            const float pcr  = rintf(pc2);
            const int   pri  = (int)prr;
            const int   pci  = (int)pcr;
            const unsigned long long key =
                ((unsigned long long)(unsigned)(r * WW + (c0 + k)) << 32)
              | (unsigned)(pri * WW + pci);
            bool  matched = false;
            float tp = 0.0f;
#pragma unroll
            for (int t = TT - 1; t >= 0; --t) {   // descending: lowest t wins
                const bool m = (key == bkey[t]);
                matched |= m;
                tp = m ? tprob[t] : tp;
            }
            const float fp = conf * conf;
            // matched => (float)pri == target r2, (float)pci == target c2
            const float cl = fabsf(pr2 - prr) + fabsf(pc2 - pcr)
                           + (conf - tp) * (conf - tp);
            sum += matched ? cl : fp;
        }
    }

    // Block reduction: WMMA wave-sum, then 8 wave partials through LDS.
    const float ws = wave_sum32(sum);
    __shared__ float lds[8];
    if ((tid & 31) == 0) lds[tid >> 5] = ws;
    __syncthreads();
    if (tid == 0) {
        float t = 0.0f;
#pragma unroll
        for (int i = 0; i < 8; ++i) t += lds[i];
        partial[blk] = t;
    }
}

// Stage 2: one block reduces all block partials, scales by 1/denom.
__global__ __launch_bounds__(256) void box_loss_stage2(
        const float* __restrict__ partial,
        float*       __restrict__ out,
        int n, float scale) {
    float s = 0.0f;
    for (int i = threadIdx.x; i < n; i += 256) s += partial[i];
    const float ws = wave_sum32(s);
    __shared__ float lds[8];
    if ((threadIdx.x & 31) == 0) lds[threadIdx.x >> 5] = ws;
    __syncthreads();
    if (threadIdx.x == 0) {
        float t = 0.0f;
#pragma unroll
        for (int i = 0; i < 8; ++i) t += lds[i];
        out[0] = t * scale;
    }
}

extern "C" void kernel_launch(void* const* d_in, const int* in_sizes, int n_in,
                              void* d_out, int out_size, void* d_ws, size_t ws_size,
                              hipStream_t stream) {
    const float* pol   = (const float*)d_in[0];   // (8, 9, 512, 512) f32
    const int*   boxes = (const int*)  d_in[1];   // (8, 16, 4) i32
    const float* probs = (const float*)d_in[2];   // (8, 16) f32
    float* out     = (float*)d_out;
    float* partial = (float*)d_ws;                // 6144 floats = 24 KB

    const int nblocks = BB * AA * (HH / 2);       // 6144
    box_loss_stage1<<<nblocks, 256, 0, stream>>>(pol, boxes, probs, partial);
    const float scale = 1.0f / (float)(BB * AA * HH * WW);
    box_loss_stage2<<<1, 256, 0, stream>>>(partial, out, nblocks, scale);
}